// MPSDist_15728170238644
// MI455X (gfx1250) — compile-verified
//
#include <hip/hip_runtime.h>
#include <hip/hip_bf16.h>
#include <cstddef>

// Problem constants
static constexpr int kR = 32;
static constexpr int kV = 32000;
static constexpr int kB = 512;
static constexpr int kT = 24;

typedef __attribute__((ext_vector_type(2))) float v2f;
typedef __attribute__((ext_vector_type(8))) float v8f;

// ---------------------------------------------------------------------------
// Kernel 0: zero the 32x32 accumulator for m in workspace
// ---------------------------------------------------------------------------
__global__ void mps_zero_m(float* __restrict__ m) {
    int i = threadIdx.x;
    if (i < kR * kR) m[i] = 0.0f;
}

// ---------------------------------------------------------------------------
// Kernel 1: m[r][s] = sum_v core[r][v][s]   (streams the whole 131 MB tensor)
// grid = (32 r-values, CHUNKS v-chunks), block = 256 (8 waves)
// float4 (b128) grid-stride loads: 512 B per wave load-instruction.
// Per-thread stride = 256*4 = 1024 floats == 0 (mod 32), so component j of
// thread t always maps to s = (4t + j) mod 32 -> trivial final reduction.
// ---------------------------------------------------------------------------
__global__ __launch_bounds__(256) void mps_reduce_m(const float* __restrict__ core,
                                                    float* __restrict__ m) {
    const int r   = blockIdx.x;
    const int t   = threadIdx.x;            // 0..255
    const int VC  = kV / gridDim.y;         // v per chunk (1280)
    const float* base = core + ((size_t)r * kV + (size_t)blockIdx.y * VC) * kR;
    const int nflt = VC * kR;               // floats in this chunk (40960)

    float4 acc = make_float4(0.f, 0.f, 0.f, 0.f);
    const float4* b4 = (const float4*)base;
    #pragma unroll 4
    for (int f = t; f < nflt / 4; f += 256) {
        float4 x = b4[f];
        acc.x += x.x; acc.y += x.y; acc.z += x.z; acc.w += x.w;
    }

    __shared__ float red[256][4];
    red[t][0] = acc.x; red[t][1] = acc.y; red[t][2] = acc.z; red[t][3] = acc.w;
    __syncthreads();

    if (t < 32) {
        // s = t; contributions come from threads tt = (t>>2) + 8*i, comp t&3
        float s = 0.f;
        #pragma unroll
        for (int i = 0; i < 32; ++i) s += red[(t >> 2) + 8 * i][t & 3];
        atomicAdd(&m[r * kR + t], s);
    }
}

// ---------------------------------------------------------------------------
// Single-wave 32x32x32 f32 GEMM using V_WMMA_F32_16X16X4_F32.
// Fragment layouts per CDNA5 ISA 7.12.2:
//   A 16x4 : VGPR q, lanes 0-15 -> (M=lane, K=q),  lanes 16-31 -> (M=lane-16, K=q+2)
//   B 4x16 : VGPR q, lanes 0-15 -> (K=q, N=lane),  lanes 16-31 -> (K=q+2, N=lane-16)
//   D 16x16: VGPR p, lanes 0-15 -> (M=p, N=lane),  lanes 16-31 -> (M=p+8, N=lane-16)
// ---------------------------------------------------------------------------
__device__ inline void matmul32_wmma(const float (*__restrict__ A)[kR],
                                     const float (*__restrict__ Bm)[kR],
                                     float (*__restrict__ D)[kR], int lane) {
    const int half = lane >> 4;      // 0 or 1
    const int l16  = lane & 15;
    #pragma unroll
    for (int ti = 0; ti < 2; ++ti) {
        #pragma unroll
        for (int tj = 0; tj < 2; ++tj) {
            v8f acc = {};
            #pragma unroll
            for (int k0 = 0; k0 < 8; ++k0) {
                const int kb = k0 * 4 + half * 2;
                v2f a, b;
                a.x = A[ti * 16 + l16][kb];
                a.y = A[ti * 16 + l16][kb + 1];
                b.x = Bm[kb][tj * 16 + l16];
                b.y = Bm[kb + 1][tj * 16 + l16];
                acc = __builtin_amdgcn_wmma_f32_16x16x4_f32(
                    false, a, false, b, (short)0, acc, false, false);
            }
            #pragma unroll
            for (int p = 0; p < 8; ++p)
                D[ti * 16 + half * 8 + p][tj * 16 + l16] = acc[p];
        }
    }
}

// ---------------------------------------------------------------------------
// Kernel 2: z = (alpha @ m^24 @ beta) * B, via matrix squaring with WMMA.
// One wave (32 threads) so EXEC is all ones for WMMA.
// ---------------------------------------------------------------------------
__global__ __launch_bounds__(32) void mps_zchain(const float* __restrict__ m,
                                                 const float* __restrict__ alpha,
                                                 const float* __restrict__ beta,
                                                 float* __restrict__ out) {
    __shared__ float L0[kR][kR];
    __shared__ float L1[kR][kR];
    __shared__ float L2[kR][kR];
    const int lane = threadIdx.x;

    // load m (row i, col lane) coalesced
    #pragma unroll 4
    for (int i = 0; i < kR; ++i) L0[i][lane] = m[i * kR + lane];

    matmul32_wmma(L0, L0, L1, lane);  // m^2
    matmul32_wmma(L1, L1, L2, lane);  // m^4
    matmul32_wmma(L2, L2, L0, lane);  // m^8   (m no longer needed)
    matmul32_wmma(L0, L0, L1, lane);  // m^16
    matmul32_wmma(L1, L0, L2, lane);  // m^24 = m^16 @ m^8

    // vz[s] = sum_r alpha[r] * m24[r][s]   (lane = s)
    float vz = 0.0f;
    #pragma unroll 8
    for (int r = 0; r < kR; ++r) vz = fmaf(alpha[r], L2[r][lane], vz);

    float prod = vz * beta[lane];
    #pragma unroll
    for (int off = 16; off > 0; off >>= 1) prod += __shfl_xor(prod, off, 32);
    if (lane == 0) out[kB] = prod * (float)kB;
}

// ---------------------------------------------------------------------------
// Kernel 3: selected chains. One wave per batch row (512 waves).
// lane = s holds state[s]; step: new[s] = sum_r state[r] * core[r][y[b,t]][s]
// Row loads are coalesced 128B L2-resident gathers; next token's 32 rows
// prefetched (lane = r) while current step computes.
// ---------------------------------------------------------------------------
__global__ __launch_bounds__(256) void mps_chains(const int* __restrict__ y,
                                                  const float* __restrict__ alpha,
                                                  const float* __restrict__ beta,
                                                  const float* __restrict__ core,
                                                  float* __restrict__ out) {
    const int lane = threadIdx.x & 31;
    const int b    = blockIdx.x * (blockDim.x >> 5) + (threadIdx.x >> 5);
    if (b >= kB) return;

    const int* yb = y + b * kT;
    float state = alpha[lane];
    int v = yb[0];
    // prefetch first matrix: lane r prefetches row r of M_v
    __builtin_prefetch(core + ((size_t)lane * kV + v) * kR, 0, 0);

    for (int t = 0; t < kT; ++t) {
        const int vn = (t + 1 < kT) ? yb[t + 1] : v;
        // prefetch next step's matrix rows while this step computes
        __builtin_prefetch(core + ((size_t)lane * kV + vn) * kR, 0, 0);

        const float* Mv = core + (size_t)v * kR;   // row r at stride kV*kR
        float ns = 0.0f;
        #pragma unroll
        for (int r = 0; r < kR; ++r) {
            const float sr = __shfl(state, r, 32);           // readlane broadcast
            ns = fmaf(sr, Mv[(size_t)r * kV * kR + lane], ns);
        }
        state = ns;
        v = vn;
    }

    float prod = state * beta[lane];
    #pragma unroll
    for (int off = 16; off > 0; off >>= 1) prod += __shfl_xor(prod, off, 32);
    if (lane == 0) out[b] = prod;
}

// ---------------------------------------------------------------------------
extern "C" void kernel_launch(void* const* d_in, const int* in_sizes, int n_in,
                              void* d_out, int out_size, void* d_ws, size_t ws_size,
                              hipStream_t stream) {
    const int*   y     = (const int*)d_in[0];
    const float* alpha = (const float*)d_in[1];
    const float* beta  = (const float*)d_in[2];
    const float* core  = (const float*)d_in[3];
    float* out = (float*)d_out;
    float* m   = (float*)d_ws;   // 32x32 f32 accumulator

    mps_zero_m<<<1, 1024, 0, stream>>>(m);
    mps_reduce_m<<<dim3(32, 25), 256, 0, stream>>>(core, m);   // 800 blocks stream 131 MB
    mps_zchain<<<1, 32, 0, stream>>>(m, alpha, beta, out);     // WMMA matrix powers
    mps_chains<<<kB / 8, 256, 0, stream>>>(y, alpha, beta, core, out);
}